// GNNFraudModel_61443802137132
// MI455X (gfx1250) — compile-verified
//
#include <hip/hip_runtime.h>
#include <math.h>

#define HID 128
#define IN_DIM 5

typedef __attribute__((ext_vector_type(2))) float v2f;
typedef __attribute__((ext_vector_type(8))) float v8f;

// ---------------------------------------------------------------- utilities
__global__ void k_fill(float* __restrict__ p, float v, int n) {
    int i = blockIdx.x * blockDim.x + threadIdx.x;
    if (i < n) p[i] = v;
}

// deg[dst] += 1 per edge (dis buffer pre-filled with 1.0 for the self loop)
__global__ void k_deg(const long long* __restrict__ ei, float* __restrict__ deg, int E) {
    int e = blockIdx.x * blockDim.x + threadIdx.x;
    if (e < E) atomicAdd(&deg[(long)ei[(long)E + e]], 1.0f);
}

__global__ void k_rsqrt(float* __restrict__ d, int n) {
    int i = blockIdx.x * blockDim.x + threadIdx.x;
    if (i < n) d[i] = rsqrtf(d[i]);   // deg >= 1 always (self loop)
}

// out[n][c] = sum_k x[n][k] * W1[k][c]   (tiny K=5 GEMM)
__global__ void k_lin1(const float* __restrict__ x, const float* __restrict__ W1,
                       float* __restrict__ out) {
    long node = blockIdx.x;
    int  c    = threadIdx.x;
    const float* xr = x + node * IN_DIM;
    float acc = 0.f;
#pragma unroll
    for (int k = 0; k < IN_DIM; ++k) acc = fmaf(xr[k], W1[k * HID + c], acc);
    out[node * HID + c] = acc;
}

// accumulator init: self-loop message (norm = dis[i]^2) + bias
__global__ void k_agg_init(const float* __restrict__ feat, const float* __restrict__ dis,
                           const float* __restrict__ bias, float* __restrict__ out) {
    long node = blockIdx.x;
    int  c    = threadIdx.x;
    float di  = dis[node];
    out[node * HID + c] = feat[node * HID + c] * di * di + bias[c];
}

// one wave per edge: lane handles 4 channels; gathers hit L2 (51MB table),
// scatter via global_atomic_add_f32 (L2-resident accumulator)
__global__ void k_agg_edges(const float* __restrict__ feat, const long long* __restrict__ ei,
                            const float* __restrict__ dis, float* __restrict__ out, int E) {
    long tid = (long)blockIdx.x * blockDim.x + threadIdx.x;
    int  e   = (int)(tid >> 5);
    if (e >= E) return;
    int  l   = (int)(tid & 31);
    long s   = (long)ei[e];
    long d   = (long)ei[(long)E + e];
    float nm = dis[s] * dis[d];
    const float4 v = ((const float4*)(feat + s * HID))[l];
    float* o = out + d * HID + l * 4;
    atomicAdd(o + 0, v.x * nm);
    atomicAdd(o + 1, v.y * nm);
    atomicAdd(o + 2, v.z * nm);
    atomicAdd(o + 3, v.w * nm);
}

__global__ void k_relu(float* __restrict__ p, long n) {
    long i = (long)blockIdx.x * blockDim.x + threadIdx.x;
    if (i < n) p[i] = fmaxf(p[i], 0.f);
}

// per-channel batch stats: block-local partial sums then 2 atomics per thread
__global__ void k_bn_stats(const float* __restrict__ h, float* __restrict__ sum,
                           float* __restrict__ sumsq, int n) {
    int c = threadIdx.x;
    float s = 0.f, sq = 0.f;
    for (int i = blockIdx.x; i < n; i += gridDim.x) {
        float v = h[(long)i * HID + c];
        s += v; sq += v * v;
    }
    atomicAdd(&sum[c], s);
    atomicAdd(&sumsq[c], sq);
}

__global__ void k_bn_final(const float* __restrict__ sum, const float* __restrict__ sumsq,
                           const float* __restrict__ gamma, const float* __restrict__ beta,
                           float* __restrict__ scale, float* __restrict__ shift, int n) {
    int c = threadIdx.x;
    float inv_n = 1.f / (float)n;
    float mean  = sum[c] * inv_n;
    float var   = sumsq[c] * inv_n - mean * mean;       // biased variance
    float is    = rsqrtf(var + 1e-5f);
    float sc    = gamma[c] * is;
    scale[c] = sc;
    shift[c] = beta[c] - mean * sc;
}

__global__ void k_bn_apply(float* __restrict__ h, const float* __restrict__ scale,
                           const float* __restrict__ shift) {
    long node = blockIdx.x;
    int  c    = threadIdx.x;
    long idx  = node * HID + c;
    h[idx] = fmaf(h[idx], scale[c], shift[c]);
}

// ---------------------------------------------------------------- WMMA GEMM
// D[N x 128] = A[N x 128] @ B[128 x 128], fp32, via V_WMMA_F32_16X16X4_F32.
// Wave owns one 16x16 tile. ISA layouts:
//   A 16x4 f32: lane m=lane&15 is row M; hi=lane>>4 selects K-half; v0/v1 = K=2hi,2hi+1
//   B 4x16:     lane n=lane&15 is col N; v0/v1 = K=2hi,2hi+1 rows
//   C/D 16x16:  VGPR r, lanes 0-15 -> M=r, lanes 16-31 -> M=r+8, N=lane&15
__global__ __launch_bounds__(256) void k_wmma_gemm(const float* __restrict__ A,
                                                   const float* __restrict__ B,
                                                   float* __restrict__ D, int n) {
    const int lane    = threadIdx.x & 31;
    const int wave    = threadIdx.x >> 5;       // 8 waves -> 8 column tiles
    const int m       = lane & 15;
    const int hi      = lane >> 4;
    const int rowBase = blockIdx.x * 16;
    const int colBase = wave * 16;

    int arow = rowBase + m;
    if (arow > n - 1) arow = n - 1;             // keep tail loads in-bounds
    const float* ap = A + (long)arow * HID + 2 * hi;
    const float* bp = B + (long)(2 * hi) * HID + colBase + m;

    v8f c = {0.f, 0.f, 0.f, 0.f, 0.f, 0.f, 0.f, 0.f};
#pragma unroll
    for (int kb = 0; kb < HID; kb += 4) {
        v2f a = *(const v2f*)(ap + kb);         // K = kb+2hi, kb+2hi+1 (8B aligned)
        v2f b;
        b.x = bp[(long)kb * HID];
        b.y = bp[(long)kb * HID + HID];
        c = __builtin_amdgcn_wmma_f32_16x16x4_f32(
                /*neg_a=*/false, a, /*neg_b=*/false, b,
                /*c_mod=*/(short)0, c, /*reuse_a=*/false, /*reuse_b=*/false);
    }
#pragma unroll
    for (int r = 0; r < 8; ++r) {
        int row = rowBase + r + 8 * hi;
        if (row < n) D[(long)row * HID + colBase + m] = c[r];
    }
}

// fused relu + (h2 @ Wfc) + bias + sigmoid; one wave32 per node
__global__ void k_final(const float* __restrict__ h2, const float* __restrict__ Wfc,
                        const float* __restrict__ bfc, float* __restrict__ out, int n) {
    long tid  = (long)blockIdx.x * blockDim.x + threadIdx.x;
    int  node = (int)(tid >> 5);
    if (node >= n) return;
    int l = (int)(tid & 31);
    float4 v = ((const float4*)(h2 + (long)node * HID))[l];
    float4 w = ((const float4*)Wfc)[l];
    float dot = fmaxf(v.x, 0.f) * w.x + fmaxf(v.y, 0.f) * w.y +
                fmaxf(v.z, 0.f) * w.z + fmaxf(v.w, 0.f) * w.w;
#pragma unroll
    for (int off = 16; off > 0; off >>= 1) dot += __shfl_xor(dot, off, 32);
    if (l == 0) out[node] = 1.f / (1.f + expf(-(dot + bfc[0])));
}

// ---------------------------------------------------------------- launcher
extern "C" void kernel_launch(void* const* d_in, const int* in_sizes, int n_in,
                              void* d_out, int out_size, void* d_ws, size_t ws_size,
                              hipStream_t stream) {
    const float*     x     = (const float*)d_in[0];
    const long long* ei    = (const long long*)d_in[1];   // int64 edge_index [2,E]
    const float*     W1    = (const float*)d_in[2];
    const float*     b1    = (const float*)d_in[3];
    const float*     W2    = (const float*)d_in[4];
    const float*     b2    = (const float*)d_in[5];
    const float*     gamma = (const float*)d_in[6];
    const float*     beta  = (const float*)d_in[7];
    const float*     Wfc   = (const float*)d_in[8];
    const float*     bfc   = (const float*)d_in[9];
    float*           out   = (float*)d_out;

    const int n = in_sizes[0] / IN_DIM;   // 100000
    const int E = in_sizes[1] / 2;        // 3200000

    // workspace carve-up (256B aligned)
    char*  ws  = (char*)d_ws;
    size_t off = 0;
    auto carve = [&](size_t bytes) -> float* {
        float* p = (float*)(ws + off);
        off += (bytes + 255) & ~(size_t)255;
        return p;
    };
    float* dis   = carve((size_t)n * 4);
    float* sum   = carve(HID * 4);
    float* sumsq = carve(HID * 4);
    float* scale = carve(HID * 4);
    float* shift = carve(HID * 4);
    float* bufA  = carve((size_t)n * HID * 4);
    float* bufB  = carve((size_t)n * HID * 4);
    (void)ws_size;

    const int  T       = 256;
    const long elems   = (long)n * HID;
    const long ethr    = (long)E * 32;
    const int  egrid   = (int)((ethr + T - 1) / T);
    const int  rowTile = (n + 15) / 16;

    // degrees -> dis = rsqrt(deg)
    k_fill <<<(n + T - 1) / T, T, 0, stream>>>(dis, 1.0f, n);           // self loop
    k_deg  <<<(E + T - 1) / T, T, 0, stream>>>(ei, dis, E);
    k_rsqrt<<<(n + T - 1) / T, T, 0, stream>>>(dis, n);

    // conv1: x@W1 -> gather/scatter -> relu
    k_lin1     <<<n, HID, 0, stream>>>(x, W1, bufB);
    k_agg_init <<<n, HID, 0, stream>>>(bufB, dis, b1, bufA);
    k_agg_edges<<<egrid, T, 0, stream>>>(bufB, ei, dis, bufA, E);
    k_relu     <<<(int)((elems + T - 1) / T), T, 0, stream>>>(bufA, elems);

    // batchnorm (training-mode batch stats)
    k_fill    <<<1, HID, 0, stream>>>(sum, 0.f, HID);
    k_fill    <<<1, HID, 0, stream>>>(sumsq, 0.f, HID);
    k_bn_stats<<<512, HID, 0, stream>>>(bufA, sum, sumsq, n);
    k_bn_final<<<1, HID, 0, stream>>>(sum, sumsq, gamma, beta, scale, shift, n);
    k_bn_apply<<<n, HID, 0, stream>>>(bufA, scale, shift);

    // conv2: WMMA fp32 GEMM h@W2, then gather/scatter
    k_wmma_gemm<<<rowTile, 256, 0, stream>>>(bufA, W2, bufB, n);
    k_agg_init <<<n, HID, 0, stream>>>(bufB, dis, b2, bufA);
    k_agg_edges<<<egrid, T, 0, stream>>>(bufB, ei, dis, bufA, E);

    // fused relu + fc + sigmoid
    k_final<<<(int)(((long)n * 32 + T - 1) / T), T, 0, stream>>>(bufA, Wfc, bfc, out, n);
    (void)out_size; (void)n_in;
}